// get_loss_6897717478086
// MI455X (gfx1250) — compile-verified
//
#include <hip/hip_runtime.h>
#include <math.h>

typedef __attribute__((ext_vector_type(2))) float v2f;
typedef __attribute__((ext_vector_type(8))) float v8f;

#define NPTS     8192
#define NBATCH   4
#define KNN      15
#define TILE     16
#define NTILES   (NPTS / TILE)           // 512 query tiles per batch
#define WPB      8                       // waves per block
#define NTHREADS (WPB * 32)              // 256
#define CHUNK_PTS   2048                 // candidate points staged per round (32 KB)
#define CHUNK_TILES (CHUNK_PTS / TILE)   // 128
#define NCHUNKS     (NPTS / CHUNK_PTS)   // 4

// One wave owns one 16-query tile:  D = A x B  (V_WMMA_F32_16X16X4_F32)
//   A (16x4) rows = candidates j: (x, y, z, |p_j|^2)   <- staged in LDS
//   B (4x16) cols = queries  i:   (-2x, -2y, -2z, 1)
//   D[j][i] = d2(i,j) - |p_i|^2   (per-column shift: top-k order safe)
// D layout (wave32): lane l -> query N = l&15, rows M = v + (l>=16 ? 8 : 0).
// Self (j==i) is the strict row minimum, so we keep top-16 and drop the
// first merged element — identical to the reference's top_k(k+1)[1:].
__global__ __launch_bounds__(NTHREADS) void knn_changerate_kernel(
    const float* __restrict__ xyz,   // [B][N][6] (pos, normal)
    float* __restrict__ out)         // [B][N]
{
    __shared__ float s_pts[CHUNK_PTS * 4];   // packed (x,y,z,|p|^2), 32 KB
    __shared__ float s_d[WPB][32][16];       // 16 KB
    __shared__ int   s_j[WPB][32][16];       // 16 KB

    const int lane = threadIdx.x & 31;
    const int wib  = threadIdx.x >> 5;
    const int wg   = blockIdx.x * WPB + wib;
    const int b    = wg / NTILES;
    const int qt   = wg - b * NTILES;
    const int qbase = qt * TILE;

    const float* bp = xyz + (size_t)b * (NPTS * 6);

    const int  sub  = lane & 15;
    const bool hi   = lane >= 16;
    const int  qg   = qbase + sub;           // this lane's query index
    const int  mrow = hi ? 8 : 0;

    // ---- B operand (queries as columns). lanes 0-15: K=0,1 -> (-2x,-2y);
    //      lanes 16-31: K=2,3 -> (-2z, 1)
    v2f bop;
    {
        const float* q = bp + (size_t)qg * 6;
        const float qx = q[0], qy = q[1], qz = q[2];
        bop.x = hi ? (-2.0f * qz) : (-2.0f * qx);
        bop.y = hi ? 1.0f         : (-2.0f * qy);
    }

    // ---- per-lane top-16, ascending
    float bd[16];
    int   bj[16];
#pragma unroll
    for (int t = 0; t < 16; ++t) { bd[t] = 1e30f; bj[t] = -1; }

    // A-operand LDS offset for this lane: point*16B + (hi ? 8B : 0)
    // -> conflict-free ds_load_b64 across all 32 lanes
    const int aoff = sub * 4 + (hi ? 2 : 0);

    for (int chunk = 0; chunk < NCHUNKS; ++chunk) {
        __syncthreads();                       // prior chunk fully consumed
        const int cbase = chunk * CHUNK_PTS;

        // ---- cooperative repack: 2048 points -> (x,y,z,|p|^2) in LDS,
        //      shared by all 8 waves of the block
#pragma unroll
        for (int r = 0; r < CHUNK_PTS / NTHREADS; ++r) {
            const int i = r * NTHREADS + threadIdx.x;
            const float* p = bp + (size_t)(cbase + i) * 6;
            const float x = p[0], y = p[1], z = p[2];
            float* dst = &s_pts[i * 4];
            dst[0] = x; dst[1] = y; dst[2] = z;
            dst[3] = x * x + y * y + z * z;
        }
        __syncthreads();

        // ---- candidate sweep, A fetch pipelined one tile ahead
        v2f aop = *(const v2f*)&s_pts[aoff];
        for (int ct = 0; ct < CHUNK_TILES; ++ct) {
            const int nt = (ct + 1) & (CHUNK_TILES - 1);     // wrap: dead value on last iter
            const v2f anext = *(const v2f*)&s_pts[nt * (TILE * 4) + aoff];

            v8f acc = {};
            acc = __builtin_amdgcn_wmma_f32_16x16x4_f32(
                /*neg_a=*/false, aop, /*neg_b=*/false, bop,
                /*c_mod=*/(short)0, acc, /*reuse_a=*/false, /*reuse_b=*/false);

            const int jrow = cbase + ct * TILE + mrow;
#pragma unroll
            for (int v = 0; v < 8; ++v) {
                const float d = acc[v];
                if (d < bd[15]) {                 // usually skipped
                    float cd = d; int cj = jrow + v;
#pragma unroll
                    for (int t = 0; t < 16; ++t) {  // sorted bubble insert
                        const float od = bd[t]; const int oj = bj[t];
                        const bool  sw = cd < od;
                        bd[t] = sw ? cd : od;  bj[t] = sw ? cj : oj;
                        cd    = sw ? od : cd;  cj    = sw ? oj : cj;
                    }
                }
            }
            aop = anext;
        }
    }

    // ---- dump lists; lane pairs (l, l+16) hold the same query
#pragma unroll
    for (int t = 0; t < 16; ++t) {
        s_d[wib][lane][t] = bd[t];
        s_j[wib][lane][t] = bj[t];
    }
    __syncthreads();

    // ---- merge two sorted 16-lists, drop first (self), fused stage 2
    if (!hi) {
        const float* ni = bp + (size_t)qg * 6 + 3;
        const float nx = ni[0], ny = ni[1], nz = ni[2];

        int pa = 0, pb = 0;
        float sum = 0.0f;
        for (int t = 0; t < KNN + 1; ++t) {
            const float da = s_d[wib][lane][pa];
            const float db = s_d[wib][lane + 16][pb];
            int j;
            if (da <= db) { j = s_j[wib][lane][pa];      ++pa; }
            else          { j = s_j[wib][lane + 16][pb]; ++pb; }
            if (t == 0) continue;                 // drop self (row minimum)

            const float* nj = bp + (size_t)j * 6 + 3;
            const float x = nj[0], y = nj[1], z = nj[2];
            const float ccx = ny * z - nz * y;
            const float ccy = nz * x - nx * z;
            const float ccz = nx * y - ny * x;
            const float cn  = sqrtf(ccx * ccx + ccy * ccy + ccz * ccz);
            const float mx = nx * x, my = ny * y, mz = nz * z;
            const float mn = sqrtf(mx * mx + my * my + mz * mz);
            sum += fminf(cn, mn);
        }
        out[(size_t)b * NPTS + qg] = sum;
    }
}

extern "C" void kernel_launch(void* const* d_in, const int* in_sizes, int n_in,
                              void* d_out, int out_size, void* d_ws, size_t ws_size,
                              hipStream_t stream) {
    const float* xyz = (const float*)d_in[0];   // [4][8192][6] float32
    // d_in[1] = num_class (unused), d_in[2] = skel_xyz (unused)
    float* out = (float*)d_out;                 // [4][8192] float32

    const int total_waves = NBATCH * NTILES;    // 2048
    dim3 grid(total_waves / WPB);               // 256 blocks
    dim3 block(NTHREADS);                       // 256 threads = 8 waves
    knn_changerate_kernel<<<grid, block, 0, stream>>>(xyz, out);
}